// DSRRNN_46179488366645
// MI455X (gfx1250) — compile-verified
//
#include <hip/hip_runtime.h>
#include <math.h>

typedef __attribute__((ext_vector_type(2))) float v2f;
typedef __attribute__((ext_vector_type(8))) float v8f;
typedef __attribute__((ext_vector_type(4))) unsigned int v4u;
typedef __attribute__((ext_vector_type(8))) int v8i;
typedef __attribute__((ext_vector_type(4))) int v4i;

#define H      256
#define IN_DIM 24
#define NOPS   12
#define MTILE  64      // batch rows per workgroup
#define KB     32      // K-block for W_hh staging

#define HP 260         // sh_h pitch (dwords): TDM pads 4 dw per 256 dw
#define WP 36          // sh_w pitch: TDM pads 4 dw per 32 dw
#define XP 24          // sh_x / sh_wi pitch (unpadded; 2-way conflicts only)
#define PP 260         // sh_wp pitch (same 256+4 padding)

// ---- Tensor Data Mover: 2D tile load global -> LDS ----------------------
// D# per CDNA5 ISA. 4-byte elements (data_size code 2).
// pad_icode: 0=2dw,1=4,...,7=256 dwords between pads; pad_acode: 0=1dw..127=128dw.
__device__ __forceinline__ void tdm_load_2d(unsigned int lds_off, const void* gptr,
                                            unsigned int tile_d0, unsigned int tile_d1,
                                            unsigned int stride0,
                                            unsigned int pad_en,
                                            unsigned int pad_icode, unsigned int pad_acode)
{
    unsigned long long ga = (unsigned long long)(uintptr_t)gptr;
    v4u g0;
    g0.x = 1u;                                                   // count=1, user mode
    g0.y = lds_off;                                              // lds_addr (bytes)
    g0.z = (unsigned int)ga;                                     // global_addr[31:0]
    g0.w = (unsigned int)((ga >> 32) & 0x1FFFFFFu) | (2u << 30); // addr[56:32] | type=2
    v8i g1;
    g1[0] = (int)((2u << 16) | (pad_en << 20) | (pad_icode << 22) | (pad_acode << 25));
    g1[1] = (int)((tile_d0 & 0xFFFFu) << 16);                    // tensor_dim0 = tile_d0
    g1[2] = (int)((tile_d0 >> 16) | ((tile_d1 & 0xFFFFu) << 16));// dim0 hi | tensor_dim1 lo
    g1[3] = (int)((tile_d1 >> 16) | (tile_d0 << 16));            // dim1 hi | tile_dim0
    g1[4] = (int)(tile_d1 & 0xFFFFu);                            // tile_dim1 (tile_dim2=0)
    g1[5] = (int)stride0;                                        // tensor_dim0_stride lo32
    g1[6] = 0;                                                   // stride0 hi | stride1 lo
    g1[7] = 0;
    v4i gz = {0, 0, 0, 0};
#if __clang_major__ >= 23
    v8i gz8 = {0, 0, 0, 0, 0, 0, 0, 0};
    __builtin_amdgcn_tensor_load_to_lds(g0, g1, gz, gz, gz8, 0);
#else
    __builtin_amdgcn_tensor_load_to_lds(g0, g1, gz, gz, 0);
#endif
}

__global__ __launch_bounds__(256) void rnn_cell_kernel(
    const float* __restrict__ x,       // [N, 24]
    const float* __restrict__ hidden,  // [N, 256]
    const float* __restrict__ W_ih,    // [256, 24]
    const float* __restrict__ b_ih,    // [256]
    const float* __restrict__ W_hh,    // [256, 256]
    const float* __restrict__ b_hh,    // [256]
    const float* __restrict__ W_proj,  // [12, 256]
    const float* __restrict__ b_proj,  // [12]
    float* __restrict__ out,           // [N, 12]
    float* __restrict__ h_out,         // [N, 256]
    int n_rows)
{
    __shared__ float sh_h[MTILE * HP];     // hidden tile, later h_new tile
    __shared__ float sh_w[2][H * WP];      // double-buffered W_hh K-blocks [n][kk]
    __shared__ float sh_x[MTILE * XP];     // x tile
    __shared__ float sh_wi[H * XP];        // W_ih [n][k]
    __shared__ float sh_wp[16 * PP];       // W_proj [16 rows][256] (rows 12..15 zero)
    __shared__ float sh_p[MTILE * 16];     // projection logits

    const int tid  = threadIdx.x;
    const int lane = tid & 31;
    const int wave = tid >> 5;
    const int rt   = wave & 3;             // row tile 0..3 (16 rows each)
    const int cb   = wave >> 2;            // col block 0..1 (128 cols each)
    const int lrow = lane & 15;
    const int khalf = (lane >> 4) << 1;    // 0 or 2 (K pair select)
    const int mhalf = (lane >> 4) * 8;     // C-layout row-half select
    const int m0 = blockIdx.x * MTILE;
    if (m0 >= n_rows) return;

    // ---- kick off all initial DMA from wave 0 (TDM ignores EXEC; 1 issue/WG) ----
    if (wave == 0) {
        tdm_load_2d((unsigned int)(uintptr_t)&sh_h[0], hidden + (size_t)m0 * H,
                    H, MTILE, H, /*pad*/1, /*256dw*/7, /*4dw*/3);
        tdm_load_2d((unsigned int)(uintptr_t)&sh_x[0], x + (size_t)m0 * IN_DIM,
                    IN_DIM, MTILE, IN_DIM, 0, 0, 0);
        tdm_load_2d((unsigned int)(uintptr_t)&sh_wi[0], W_ih,
                    IN_DIM, H, IN_DIM, 0, 0, 0);
        tdm_load_2d((unsigned int)(uintptr_t)&sh_wp[0], W_proj,
                    H, NOPS, H, /*pad*/1, 7, 3);
        tdm_load_2d((unsigned int)(uintptr_t)&sh_w[0][0], W_hh,          // K-block 0
                    KB, H, H, /*pad*/1, /*32dw*/4, /*4dw*/3);
    }
    // zero padding rows 12..15 of W_proj tile (disjoint from TDM writes)
    for (int idx = tid; idx < 4 * PP; idx += 256) sh_wp[NOPS * PP + idx + (16 - NOPS - 4) * PP] = 0.f;

    // ---- accumulators: init with biases (no LDS dependence) ----
    v8f acc[8];
#pragma unroll
    for (int t = 0; t < 8; ++t) {
        int n = cb * 128 + t * 16 + lrow;
        float b = b_ih[n] + b_hh[n];
        acc[t] = (v8f){b, b, b, b, b, b, b, b};
    }

    if (wave == 0) __builtin_amdgcn_s_wait_tensorcnt(0);
    __syncthreads();                        // all initial tiles resident

    // ---- input-to-hidden GEMM: K = 24 ----
#pragma unroll
    for (int k0 = 0; k0 < IN_DIM; k0 += 4) {
        v2f a = *(const v2f*)&sh_x[(rt * 16 + lrow) * XP + k0 + khalf];
#pragma unroll
        for (int t = 0; t < 8; ++t) {
            v2f b = *(const v2f*)&sh_wi[(cb * 128 + t * 16 + lrow) * XP + k0 + khalf];
            acc[t] = __builtin_amdgcn_wmma_f32_16x16x4_f32(
                false, a, false, b, (short)0, acc[t], false, false);
        }
    }

    // ---- hidden-to-hidden GEMM: K = 256, double-buffered TDM K-blocks ----
    for (int kb = 0; kb < H; kb += KB) {
        const int buf = (kb >> 5) & 1;
        if (kb > 0) {
            if (wave == 0) __builtin_amdgcn_s_wait_tensorcnt(0);  // block kb arrived
            __syncthreads();            // visible to all; buf^1 fully consumed
        }
        if (wave == 0 && kb + KB < H)
            tdm_load_2d((unsigned int)(uintptr_t)&sh_w[buf ^ 1][0],
                        W_hh + (kb + KB), KB, H, H, 1, 4, 3);     // overlap with WMMA

        const float* wblk = sh_w[buf];
#pragma unroll
        for (int kk = 0; kk < KB; kk += 4) {
            v2f a = *(const v2f*)&sh_h[(rt * 16 + lrow) * HP + kb + kk + khalf];
#pragma unroll
            for (int t = 0; t < 8; ++t) {
                v2f b = *(const v2f*)&wblk[(cb * 128 + t * 16 + lrow) * WP + kk + khalf];
                acc[t] = __builtin_amdgcn_wmma_f32_16x16x4_f32(
                    false, a, false, b, (short)0, acc[t], false, false);
            }
        }
    }

    // ---- tanh nonlinearity ----
#pragma unroll
    for (int t = 0; t < 8; ++t)
#pragma unroll
        for (int e = 0; e < 8; ++e)
            acc[t][e] = tanhf(acc[t][e]);

    __syncthreads();   // everyone done reading sh_h as "hidden"

    // ---- write h_new to global + LDS (C layout: VGPR e -> M = e + 8*(lane>>4)) ----
#pragma unroll
    for (int t = 0; t < 8; ++t) {
        int n = cb * 128 + t * 16 + lrow;
#pragma unroll
        for (int e = 0; e < 8; ++e) {
            int ml = rt * 16 + mhalf + e;
            float v = acc[t][e];
            sh_h[ml * HP + n] = v;
            h_out[(size_t)(m0 + ml) * H + n] = v;
        }
    }
    __syncthreads();

    // ---- projection GEMM on WMMA: waves 0..3, one 16x16 logits tile each ----
    if (wave < 4) {
        float pb = (lrow < NOPS) ? b_proj[lrow] : 0.f;
        v8f pacc = (v8f){pb, pb, pb, pb, pb, pb, pb, pb};
#pragma unroll 16
        for (int kk = 0; kk < H; kk += 4) {
            v2f a = *(const v2f*)&sh_h[(rt * 16 + lrow) * HP + kk + khalf];   // h_new rows
            v2f b = *(const v2f*)&sh_wp[lrow * PP + kk + khalf];              // W_proj rows
            pacc = __builtin_amdgcn_wmma_f32_16x16x4_f32(
                false, a, false, b, (short)0, pacc, false, false);
        }
#pragma unroll
        for (int e = 0; e < 8; ++e)
            sh_p[(rt * 16 + mhalf + e) * 16 + lrow] = pacc[e];
    }
    __syncthreads();

    // ---- softmax over 12 ops, one row per thread ----
    if (tid < MTILE) {
        float v[NOPS];
        float mx = -INFINITY;
#pragma unroll
        for (int o = 0; o < NOPS; ++o) { v[o] = sh_p[tid * 16 + o]; mx = fmaxf(mx, v[o]); }
        float sum = 0.f;
#pragma unroll
        for (int o = 0; o < NOPS; ++o) { v[o] = __expf(v[o] - mx); sum += v[o]; }
        float inv = 1.0f / sum;
        float* orow = out + (size_t)(m0 + tid) * NOPS;
#pragma unroll
        for (int o = 0; o < NOPS; ++o) orow[o] = v[o] * inv;
    }
}

extern "C" void kernel_launch(void* const* d_in, const int* in_sizes, int n_in,
                              void* d_out, int out_size, void* d_ws, size_t ws_size,
                              hipStream_t stream) {
    const float* x      = (const float*)d_in[0];
    const float* hidden = (const float*)d_in[1];
    const float* W_ih   = (const float*)d_in[2];
    const float* b_ih   = (const float*)d_in[3];
    const float* W_hh   = (const float*)d_in[4];
    const float* b_hh   = (const float*)d_in[5];
    const float* W_proj = (const float*)d_in[6];
    const float* b_proj = (const float*)d_in[7];

    const int N = in_sizes[1] / H;              // hidden is [N, 256]
    float* out   = (float*)d_out;               // [N, 12] first
    float* h_out = out + (size_t)N * NOPS;      // then [N, 256]

    dim3 grid((N + MTILE - 1) / MTILE), block(256);
    rnn_cell_kernel<<<grid, block, 0, stream>>>(
        x, hidden, W_ih, b_ih, W_hh, b_hh, W_proj, b_proj, out, h_out, N);
}